// GeneralizingProjection_27668179321271
// MI455X (gfx1250) — compile-verified
//
#include <hip/hip_runtime.h>
#include <stdint.h>

#define NBITS 16
#define TABLE_SIZE 32768                      // 2^(NBITS-1)
#define WORDS_PER_TABLE (TABLE_SIZE / 32)     // 1024
#define PACKED_WORDS (NBITS * WORDS_PER_TABLE)// 16384 dwords
#define PACKED_BYTES (PACKED_WORDS * 4)       // 65536 bytes

typedef int   vi4 __attribute__((ext_vector_type(4)));
typedef float vf4 __attribute__((ext_vector_type(4)));

// ---------------------------------------------------------------------------
// Kernel 1: bit-pack the 16 x 32768 float (0/1) tables into 64KB of dwords.
// word w (0..16383): pos = w>>10, entries [ (w&1023)*32 .. +31 ], bit j = entry j.
// ---------------------------------------------------------------------------
__global__ __launch_bounds__(256)
void pack_tables_kernel(const float* __restrict__ tables,
                        uint32_t* __restrict__ packed) {
    int w = blockIdx.x * blockDim.x + threadIdx.x;
    if (w >= PACKED_WORDS) return;
    int pos  = w >> 10;
    int base = (w & 1023) << 5;
    const float* t = tables + (size_t)pos * TABLE_SIZE + base;
    uint32_t bw = 0;
#pragma unroll
    for (int j = 0; j < 32; ++j) {
        float f = t[j];
        bw |= (f != 0.0f ? 1u : 0u) << j;
    }
    packed[w] = bw;
}

// ---------------------------------------------------------------------------
// Kernel 2 (main): stage the 64KB bit table into LDS via CDNA5 async-to-LDS
// (global_load_async_to_lds_b128, ASYNCcnt), then stream rows: pack the 16-bit
// LSB-first mask m; for each position p the context index is just the low p
// bits of m; lookup = LDS bit test; out bit = b ^ flip, emitted MSB-first.
// ---------------------------------------------------------------------------
__global__ __launch_bounds__(256)
void bitmapper_lds_kernel(const int* __restrict__ bits,
                          const uint32_t* __restrict__ packed,
                          float* __restrict__ out,
                          int batch) {
    __shared__ uint32_t sTab[PACKED_WORDS];   // 64 KB
    const int tid = threadIdx.x;

    // --- async copy global -> LDS: 64B per thread = 4 x B128, at a 256B thread stride ---------
    {
        unsigned ldsBase = (unsigned)(uintptr_t)(&sTab[0]) + (unsigned)tid * 256u;
        uint64_t gBase   = (uint64_t)(uintptr_t)((const char*)packed + (size_t)tid * 256);
#pragma unroll
        for (int i = 0; i < 4; ++i) {
            unsigned la = ldsBase + (unsigned)(i * 16);
            uint64_t ga = gBase + (uint64_t)(i * 16);
            asm volatile("global_load_async_to_lds_b128 %0, %1, off"
                         :: "v"(la), "v"(ga) : "memory");
        }
        asm volatile("s_wait_asynccnt 0" ::: "memory");
        __syncthreads();
    }

    const unsigned gid      = blockIdx.x * blockDim.x + tid;
    const unsigned gridSize = gridDim.x * blockDim.x;

    for (size_t r = gid; r < (size_t)batch; r += gridSize) {
        // 64B coalesced non-temporal load of one row (16 x int32, MSB-first)
        const vi4* bp = (const vi4*)(bits + r * NBITS);
        vi4 b0 = __builtin_nontemporal_load(&bp[0]);
        vi4 b1 = __builtin_nontemporal_load(&bp[1]);
        vi4 b2 = __builtin_nontemporal_load(&bp[2]);
        vi4 b3 = __builtin_nontemporal_load(&bp[3]);
        const int v[NBITS] = { b0.x, b0.y, b0.z, b0.w,  b1.x, b1.y, b1.z, b1.w,
                               b2.x, b2.y, b2.z, b2.w,  b3.x, b3.y, b3.z, b3.w };

        // m bit p (LSB-first) = bits[NBITS-1-p]
        unsigned m = 0;
#pragma unroll
        for (int k = 0; k < NBITS; ++k)
            m |= ((unsigned)v[k] & 1u) << (NBITS - 1 - k);

        float o[NBITS];
#pragma unroll
        for (int p = 0; p < NBITS; ++p) {
            unsigned b    = (m >> p) & 1u;
            unsigned idx  = m & ((1u << p) - 1u);          // exclusive prefix ctx
            unsigned word = sTab[(p << 10) + (idx >> 5)];  // ds_load_b32
            unsigned flip = (word >> (idx & 31u)) & 1u;
            o[NBITS - 1 - p] = (float)(b ^ flip);          // XOR, back to MSB-first
        }

        vf4* op = (vf4*)(out + r * NBITS);
        vf4 o0 = { o[0],  o[1],  o[2],  o[3]  };
        vf4 o1 = { o[4],  o[5],  o[6],  o[7]  };
        vf4 o2 = { o[8],  o[9],  o[10], o[11] };
        vf4 o3 = { o[12], o[13], o[14], o[15] };
        __builtin_nontemporal_store(o0, &op[0]);
        __builtin_nontemporal_store(o1, &op[1]);
        __builtin_nontemporal_store(o2, &op[2]);
        __builtin_nontemporal_store(o3, &op[3]);
    }
}

// ---------------------------------------------------------------------------
// Fallback (ws too small): gather flips straight from the float tables (L2).
// ---------------------------------------------------------------------------
__global__ __launch_bounds__(256)
void bitmapper_direct_kernel(const int* __restrict__ bits,
                             const float* __restrict__ tables,
                             float* __restrict__ out,
                             int batch) {
    const unsigned gid      = blockIdx.x * blockDim.x + threadIdx.x;
    const unsigned gridSize = gridDim.x * blockDim.x;
    for (size_t r = gid; r < (size_t)batch; r += gridSize) {
        const vi4* bp = (const vi4*)(bits + r * NBITS);
        vi4 b0 = __builtin_nontemporal_load(&bp[0]);
        vi4 b1 = __builtin_nontemporal_load(&bp[1]);
        vi4 b2 = __builtin_nontemporal_load(&bp[2]);
        vi4 b3 = __builtin_nontemporal_load(&bp[3]);
        const int v[NBITS] = { b0.x, b0.y, b0.z, b0.w,  b1.x, b1.y, b1.z, b1.w,
                               b2.x, b2.y, b2.z, b2.w,  b3.x, b3.y, b3.z, b3.w };
        unsigned m = 0;
#pragma unroll
        for (int k = 0; k < NBITS; ++k)
            m |= ((unsigned)v[k] & 1u) << (NBITS - 1 - k);
        float o[NBITS];
#pragma unroll
        for (int p = 0; p < NBITS; ++p) {
            unsigned b   = (m >> p) & 1u;
            unsigned idx = m & ((1u << p) - 1u);
            float f = tables[((size_t)p << 15) + idx];
            unsigned flip = (f != 0.0f) ? 1u : 0u;
            o[NBITS - 1 - p] = (float)(b ^ flip);
        }
        vf4* op = (vf4*)(out + r * NBITS);
        vf4 o0 = { o[0],  o[1],  o[2],  o[3]  };
        vf4 o1 = { o[4],  o[5],  o[6],  o[7]  };
        vf4 o2 = { o[8],  o[9],  o[10], o[11] };
        vf4 o3 = { o[12], o[13], o[14], o[15] };
        __builtin_nontemporal_store(o0, &op[0]);
        __builtin_nontemporal_store(o1, &op[1]);
        __builtin_nontemporal_store(o2, &op[2]);
        __builtin_nontemporal_store(o3, &op[3]);
    }
}

extern "C" void kernel_launch(void* const* d_in, const int* in_sizes, int n_in,
                              void* d_out, int out_size, void* d_ws, size_t ws_size,
                              hipStream_t stream) {
    const int*   bits   = (const int*)d_in[0];     // (BATCH, 16) int32
    const float* tables = (const float*)d_in[1];   // (16, 32768) float32
    float*       out    = (float*)d_out;           // (BATCH, 16) float32
    const int batch = in_sizes[0] / NBITS;

    const int threads = 256;
    const int blocks  = 2048;                      // 8 rows/thread at BATCH=4M

    if (ws_size >= (size_t)PACKED_BYTES) {
        uint32_t* packed = (uint32_t*)d_ws;
        pack_tables_kernel<<<PACKED_WORDS / 256, 256, 0, stream>>>(tables, packed);
        bitmapper_lds_kernel<<<blocks, threads, 0, stream>>>(bits, packed, out, batch);
    } else {
        bitmapper_direct_kernel<<<blocks, threads, 0, stream>>>(bits, tables, out, batch);
    }
}